// HyperGATConv_34883724378624
// MI455X (gfx1250) — compile-verified
//
#include <hip/hip_runtime.h>
#include <math.h>

// ---------------------------------------------------------------------------
// HyperGAT conv for MI455X (gfx1250, wave32).
//   h       = x @ theta + bias                 (WMMA f32 16x16x4 GEMM)
//   e_feat  = segmean_e(h[node_idx])           (atomic scatter, wave/pair)
//   h2      = segmean_v(e_feat[edge_idx])
//   e_feat2 = segmean_e(h2[node_idx])
//   w       = sigmoid(e_feat2 @ att_w + att_b) (wave-per-edge dot + shfl)
//   out     = segsum_v(w*e_feat2[edge_idx]) / max(segsum_v(w), 1e-12)
// ---------------------------------------------------------------------------

typedef __attribute__((ext_vector_type(2))) float v2f;
typedef __attribute__((ext_vector_type(8))) float v8f;

static constexpr int D = 128;     // feature dim (D_IN == D_OUT)

// ---------------------------------------------------------------------------
__global__ void zero_f32(float* __restrict__ p, long n) {
  long i = (long)blockIdx.x * blockDim.x + threadIdx.x;
  if (i < n) p[i] = 0.0f;
}

// ---------------------------------------------------------------------------
// GEMM: h[rowbase..rowbase+15][0..127] = x @ theta + bias, fp32 WMMA.
// Block = 256 threads = 8 waves; wave w computes the 16x16 tile at cols w*16.
// theta (128x128 f32 = 64 KB) staged in LDS once per block; x tile (16x128)
// staged with stride-132 padding so A-fragment reads don't bank-conflict
// (stride 128 would land all 16 lanes on the same bank: 128 % 64 == 0).
__global__ __launch_bounds__(256) void gemm_bias_wmma(
    const float* __restrict__ x, const float* __restrict__ theta,
    const float* __restrict__ bias, float* __restrict__ h, int nrows) {
  __shared__ __align__(16) float sT[D * D];     // theta row-major [k][n]
  __shared__ __align__(16) float sX[16 * 132];  // 16 x rows, padded stride

  const int tid = threadIdx.x;
  const int rowbase = blockIdx.x * 16;

  // cooperative load: theta -> LDS (float4, fully coalesced)
  {
    const float4* src = reinterpret_cast<const float4*>(theta);
    float4* dst = reinterpret_cast<float4*>(sT);
    for (int i = tid; i < D * D / 4; i += 256) dst[i] = src[i];
  }
  // cooperative load: 16 rows of x -> LDS (zero-pad past nrows)
  for (int i = tid; i < 16 * (D / 4); i += 256) {
    const int r = i / (D / 4);
    const int c = i % (D / 4);
    float4 v = make_float4(0.f, 0.f, 0.f, 0.f);
    const int row = rowbase + r;
    if (row < nrows) v = reinterpret_cast<const float4*>(x)[row * (D / 4) + c];
    float* dp = &sX[r * 132 + c * 4];
    dp[0] = v.x; dp[1] = v.y; dp[2] = v.z; dp[3] = v.w;
  }
  __syncthreads();

  const int lane  = tid & 31;
  const int n0    = (tid >> 5) * 16;    // wave -> 16-col tile
  const int mlo   = lane & 15;          // A: M index / B,C,D: N index
  const int khalf = (lane >> 4) * 2;    // A/B ISA layout: lanes 0-15 K={0,1},
                                        //                 lanes 16-31 K={2,3}
  v8f acc = {};
#pragma unroll
  for (int k0 = 0; k0 < D; k0 += 4) {
    v2f a, b;
    const float2 av =
        *reinterpret_cast<const float2*>(&sX[mlo * 132 + k0 + khalf]);
    a.x = av.x;
    a.y = av.y;
    b.x = sT[(k0 + khalf) * D + n0 + mlo];
    b.y = sT[(k0 + khalf + 1) * D + n0 + mlo];
    // D = A(16x4,f32) * B(4x16,f32) + C  -> v_wmma_f32_16x16x4_f32
    acc = __builtin_amdgcn_wmma_f32_16x16x4_f32(false, a, false, b, (short)0,
                                                acc, false, false);
  }

  // C/D layout: VGPR v -> M = v (lanes 0-15) or v+8 (lanes 16-31), N = lane&15
  const int nIdx = n0 + mlo;
  const float bv = bias[nIdx];
  const int mhi = (lane >> 4) * 8;
#pragma unroll
  for (int v = 0; v < 8; ++v) {
    const int row = rowbase + v + mhi;
    if (row < nrows) h[(size_t)row * D + nIdx] = acc[v] + bv;
  }
}

// ---------------------------------------------------------------------------
// One wave per incidence pair: gather src[gidx[i]] (128 f32 = 32 lanes x f4),
// atomically accumulate into sum[sidx[i]]; lane 0 bumps the segment count.
__global__ __launch_bounds__(256) void scatter_accum(
    const float4* __restrict__ src, const int* __restrict__ gidx,
    const int* __restrict__ sidx, float* __restrict__ sum,
    float* __restrict__ cnt, int nnz) {
  const long gid = (long)blockIdx.x * blockDim.x + threadIdx.x;
  const int pair = (int)(gid >> 5);
  const int lane = (int)(gid & 31);
  if (pair >= nnz) return;
  const int s = gidx[pair];
  const int d = sidx[pair];
  const float4 v = src[(size_t)s * 32 + lane];
  float* dp = sum + (size_t)d * D + lane * 4;
  atomicAdd(dp + 0, v.x);
  atomicAdd(dp + 1, v.y);
  atomicAdd(dp + 2, v.z);
  atomicAdd(dp + 3, v.w);
  if (lane == 0 && cnt != nullptr) atomicAdd(cnt + d, 1.0f);
}

// feat[r][:] /= max(cnt[r], 1)   (nvec = rows * 32 float4s)
__global__ void div_by_cnt(float4* __restrict__ feat,
                           const float* __restrict__ cnt, long nvec) {
  const long i = (long)blockIdx.x * blockDim.x + threadIdx.x;
  if (i >= nvec) return;
  const float c = fmaxf(cnt[i >> 5], 1.0f);
  float4 v = feat[i];
  v.x /= c; v.y /= c; v.z /= c; v.w /= c;
  feat[i] = v;
}

// ---------------------------------------------------------------------------
// One wave per hyperedge: w[e] = sigmoid(dot(e_feat2[e], att_w) + att_b)
__global__ __launch_bounds__(256) void edge_attention(
    const float4* __restrict__ efeat, const float* __restrict__ att_w,
    const float* __restrict__ att_b, float* __restrict__ w, int E) {
  const long gid = (long)blockIdx.x * blockDim.x + threadIdx.x;
  const int e = (int)(gid >> 5);
  const int lane = (int)(gid & 31);
  if (e >= E) return;
  const float4 v = efeat[(size_t)e * 32 + lane];
  const float4 aw = reinterpret_cast<const float4*>(att_w)[lane];
  float s = v.x * aw.x + v.y * aw.y + v.z * aw.z + v.w * aw.w;
#pragma unroll
  for (int off = 16; off > 0; off >>= 1) s += __shfl_xor(s, off, 32);
  if (lane == 0) {
    const float z = s + att_b[0];
    w[e] = 1.0f / (1.0f + expf(-z));
  }
}

// One wave per pair: num[n] += w[e] * e_feat2[e]; den[n] += w[e]
__global__ __launch_bounds__(256) void scatter_weighted(
    const float4* __restrict__ efeat, const float* __restrict__ w,
    const int* __restrict__ node_idx, const int* __restrict__ edge_idx,
    float* __restrict__ num, float* __restrict__ den, int nnz) {
  const long gid = (long)blockIdx.x * blockDim.x + threadIdx.x;
  const int pair = (int)(gid >> 5);
  const int lane = (int)(gid & 31);
  if (pair >= nnz) return;
  const int e = edge_idx[pair];
  const int nd = node_idx[pair];
  const float wv = w[e];
  const float4 v = efeat[(size_t)e * 32 + lane];
  float* dp = num + (size_t)nd * D + lane * 4;
  atomicAdd(dp + 0, wv * v.x);
  atomicAdd(dp + 1, wv * v.y);
  atomicAdd(dp + 2, wv * v.z);
  atomicAdd(dp + 3, wv * v.w);
  if (lane == 0) atomicAdd(den + nd, wv);
}

__global__ void final_div(float4* __restrict__ out,
                          const float* __restrict__ den, long nvec) {
  const long i = (long)blockIdx.x * blockDim.x + threadIdx.x;
  if (i >= nvec) return;
  const float c = fmaxf(den[i >> 5], 1e-12f);
  float4 v = out[i];
  v.x /= c; v.y /= c; v.z /= c; v.w /= c;
  out[i] = v;
}

// ---------------------------------------------------------------------------
static inline int nblk(long n, int t) { return (int)((n + t - 1) / t); }

extern "C" void kernel_launch(void* const* d_in, const int* in_sizes, int n_in,
                              void* d_out, int out_size, void* d_ws,
                              size_t ws_size, hipStream_t stream) {
  const float* x      = (const float*)d_in[0];
  const float* theta  = (const float*)d_in[1];
  const float* bias   = (const float*)d_in[2];
  const float* att_w  = (const float*)d_in[3];
  const float* att_b  = (const float*)d_in[4];
  const int* node_idx = (const int*)d_in[5];
  const int* edge_idx = (const int*)d_in[6];

  const int N   = in_sizes[0] / D;   // 100000
  const int NNZ = in_sizes[5];       // 800000
  const int E   = 25000;             // reference constant (not in input shapes)

  // workspace layout (floats): ~65 MB total
  float* ws   = (float*)d_ws;
  float* h    = ws;                       // [N*128]  h, then reused as n_sum/h2
  float* ebuf = h + (size_t)N * D;        // [E*128]  e_sum/e_feat, then e_feat2
  float* ecnt = ebuf + (size_t)E * D;     // [E]
  float* ncnt = ecnt + E;                 // [N]
  float* wgt  = ncnt + N;                 // [E]
  float* den  = wgt + E;                  // [N]
  float* out  = (float*)d_out;            // [N*128]  num, then final result

  const int T = 256;

  // 1) h = x @ theta + bias   (WMMA fp32)
  gemm_bias_wmma<<<(N + 15) / 16, T, 0, stream>>>(x, theta, bias, h, N);

  // 2) e_feat = segment_mean(h[node_idx], edge_idx, E)
  zero_f32<<<nblk((long)E * D, T), T, 0, stream>>>(ebuf, (long)E * D);
  zero_f32<<<nblk(E, T), T, 0, stream>>>(ecnt, E);
  scatter_accum<<<nblk((long)NNZ * 32, T), T, 0, stream>>>(
      (const float4*)h, node_idx, edge_idx, ebuf, ecnt, NNZ);
  div_by_cnt<<<nblk((long)E * 32, T), T, 0, stream>>>((float4*)ebuf, ecnt,
                                                      (long)E * 32);

  // 3) h2 = segment_mean(e_feat[edge_idx], node_idx, N)   (reuses h buffer)
  zero_f32<<<nblk((long)N * D, T), T, 0, stream>>>(h, (long)N * D);
  zero_f32<<<nblk(N, T), T, 0, stream>>>(ncnt, N);
  scatter_accum<<<nblk((long)NNZ * 32, T), T, 0, stream>>>(
      (const float4*)ebuf, edge_idx, node_idx, h, ncnt, NNZ);
  div_by_cnt<<<nblk((long)N * 32, T), T, 0, stream>>>((float4*)h, ncnt,
                                                      (long)N * 32);

  // 4) e_feat2 = segment_mean(h2[node_idx], edge_idx, E)  (ecnt reused)
  zero_f32<<<nblk((long)E * D, T), T, 0, stream>>>(ebuf, (long)E * D);
  scatter_accum<<<nblk((long)NNZ * 32, T), T, 0, stream>>>(
      (const float4*)h, node_idx, edge_idx, ebuf, nullptr, NNZ);
  div_by_cnt<<<nblk((long)E * 32, T), T, 0, stream>>>((float4*)ebuf, ecnt,
                                                      (long)E * 32);

  // 5) w = sigmoid(e_feat2 @ att_w + att_b)
  edge_attention<<<nblk((long)E * 32, T), T, 0, stream>>>(
      (const float4*)ebuf, att_w, att_b, wgt, E);

  // 6) out = segsum(w * e_feat2[edge_idx]) / max(segsum(w), 1e-12)
  zero_f32<<<nblk((long)N * D, T), T, 0, stream>>>(out, (long)N * D);
  zero_f32<<<nblk(N, T), T, 0, stream>>>(den, N);
  scatter_weighted<<<nblk((long)NNZ * 32, T), T, 0, stream>>>(
      (const float4*)ebuf, wgt, node_idx, edge_idx, out, den, NNZ);
  final_div<<<nblk((long)N * 32, T), T, 0, stream>>>((float4*)out, den,
                                                     (long)N * 32);
}